// PointWarping2_71863392797316
// MI455X (gfx1250) — compile-verified
//
#include <hip/hip_runtime.h>

// PointWarping2 for MI455X (gfx1250).
// B=2, C=3, N1=N2=8192. Compute-bound: 134M pair interactions, ~1.5MB data.
// Strategy: V_WMMA_F32_16X16X4_F32 computes log2(w) for a 16x16 pair tile in
// one instruction; v_exp_f32 per element; fp32 VALU accumulation (robust for
// tiny weights); cross-lane butterfly reduction; prefetch of source stream.

typedef float v2f __attribute__((ext_vector_type(2)));
typedef float v8f __attribute__((ext_vector_type(8)));

#define PW_B  2
#define PW_N1 8192
#define PW_N2 8192
#define PW_LOG2E 1.4426950408889634f

// ---------------------------------------------------------------------------
// Prep kernel: per source n, precompute
//   g = (2yx, 2yy, 2yz, -|y|^2) * log2e/s^2   (WMMA B-column payload)
//   f = (f1x, f1y, f1z, 0)                     (values to regress)
// packed as two float4s in workspace. y = x1 + f1.
// ---------------------------------------------------------------------------
__global__ void pw2_prep(const float* __restrict__ xyz1,
                         const float* __restrict__ flow1,
                         const int*   __restrict__ resol,
                         float4*      __restrict__ ws) {
  int i = blockIdx.x * blockDim.x + threadIdx.x;
  if (i >= PW_B * PW_N1) return;
  int b = i / PW_N1;
  int n = i - b * PW_N1;
  float s = 1.0f * (float)resol[0];          // INITIAL_RADIUS * resol_factor
  float k = PW_LOG2E / (s * s);
  const float* x = xyz1  + (size_t)b * 3 * PW_N1 + n;
  const float* f = flow1 + (size_t)b * 3 * PW_N1 + n;
  float fx = f[0], fy = f[PW_N1], fz = f[2 * PW_N1];
  float yx = x[0]        + fx;
  float yy = x[PW_N1]    + fy;
  float yz = x[2*PW_N1]  + fz;
  float4 g = make_float4(2.f * yx * k, 2.f * yy * k, 2.f * yz * k,
                         -(yx * yx + yy * yy + yz * yz) * k);
  float4 ff = make_float4(fx, fy, fz, 0.f);
  ws[2 * (size_t)i]     = g;
  ws[2 * (size_t)i + 1] = ff;
}

// ---------------------------------------------------------------------------
// Main kernel: one wave32 owns a tile of 16 queries; loops over all 512
// source tiles. Per tile: 1 WMMA (f32 16x16x4) + 8 exp + 32 FMA per lane.
// ---------------------------------------------------------------------------
__global__ __launch_bounds__(128) void pw2_main(
    const float*  __restrict__ xyz2,
    const int*    __restrict__ resol,
    const float4* __restrict__ ws,
    float*        __restrict__ out) {
  const int lane  = threadIdx.x & 31;
  const int wave  = threadIdx.x >> 5;
  const int tile  = blockIdx.x * 4 + wave;         // 0 .. B*N2/16 - 1
  const int tpb   = PW_N2 / 16;                    // tiles per batch = 512
  const int b     = tile / tpb;
  const int qbase = (tile - b * tpb) * 16;
  const int h     = lane >> 4;                     // lane half (0/1)
  const int m     = lane & 15;                     // query/source column slot

  const float s = (float)resol[0];
  const float k = PW_LOG2E / (s * s);

  // Query coords for row m (every lane loads all 3 components).
  const float* xq = xyz2 + (size_t)b * 3 * PW_N2;
  const float qx = xq[qbase + m];
  const float qy = xq[PW_N2 + qbase + m];
  const float qz = xq[2 * PW_N2 + qbase + m];
  const float q2 = (qx * qx + qy * qy + qz * qz) * k;

  // A matrix (16x4 f32): lanes 0-15 hold K=0,1 ; lanes 16-31 hold K=2,3.
  v2f A;
  A.x = h ? qz   : qx;
  A.y = h ? 1.0f : qy;

  // C matrix preload: C[m,n] = -|q_m|^2 * log2e/s^2 (constant across n).
  // C/D layout: lane in half h, VGPR v  <->  row m = v + 8h.
  v8f C;
#pragma unroll
  for (int v = 0; v < 8; ++v)
    C[v] = -__shfl(q2, v + 8 * h, 32);

  // Per-lane accumulators: this lane's column n contributes to 8 rows.
  float ax[8], ay[8], az[8], aw[8];
#pragma unroll
  for (int v = 0; v < 8; ++v) { ax[v] = 0.f; ay[v] = 0.f; az[v] = 0.f; aw[v] = 0.f; }

  // Source stream: lane handles source n = j*16 + m (lanes i and i+16 share n).
  const float4* src = ws + (size_t)b * PW_N1 * 2 + 2 * m;

#pragma unroll 2
  for (int j = 0; j < PW_N1 / 16; ++j) {
    float4 g = src[j * 32];          // (2yx,2yy,2yz,-|y|^2)*log2e/s^2
    float4 f = src[j * 32 + 1];      // (f1x,f1y,f1z,0)
    __builtin_prefetch(&src[j * 32 + 8 * 32], 0, 0);  // 8 tiles ahead

    // B matrix (4x16 f32): lanes 0-15 hold K=0,1 ; lanes 16-31 hold K=2,3.
    v2f Bm;
    Bm.x = h ? g.z : g.x;
    Bm.y = h ? g.w : g.y;

    // S[m,n] = (2 q.y - |y|^2 - |q|^2) * log2e/s^2 = log2(exp(-d2/s^2))
    v8f S = __builtin_amdgcn_wmma_f32_16x16x4_f32(
        false, A, false, Bm, (short)0, C, false, false);

#pragma unroll
    for (int v = 0; v < 8; ++v) {
      float w = __builtin_amdgcn_exp2f(S[v]);   // Gaussian kernel weight
      ax[v] = fmaf(w, f.x, ax[v]);
      ay[v] = fmaf(w, f.y, ay[v]);
      az[v] = fmaf(w, f.z, az[v]);
      aw[v] += w;
    }
  }

  // Reduce over the 16 columns held by each lane half (xor masks stay in half).
#pragma unroll
  for (int d = 1; d < 16; d <<= 1) {
#pragma unroll
    for (int v = 0; v < 8; ++v) {
      ax[v] += __shfl_xor(ax[v], d, 32);
      ay[v] += __shfl_xor(ay[v], d, 32);
      az[v] += __shfl_xor(az[v], d, 32);
      aw[v] += __shfl_xor(aw[v], d, 32);
    }
  }

  // Row v+8h is complete in every lane of half h; lane (m==v) writes it.
  float* ob = out + (size_t)b * 3 * PW_N2;
#pragma unroll
  for (int v = 0; v < 8; ++v) {
    if (m == v) {
      int q = qbase + v + 8 * h;
      float inv = 1.0f / aw[v];
      ob[q]             = xq[q]             - ax[v] * inv;
      ob[PW_N2 + q]     = xq[PW_N2 + q]     - ay[v] * inv;
      ob[2 * PW_N2 + q] = xq[2 * PW_N2 + q] - az[v] * inv;
    }
  }
}

// ---------------------------------------------------------------------------
extern "C" void kernel_launch(void* const* d_in, const int* in_sizes, int n_in,
                              void* d_out, int out_size, void* d_ws, size_t ws_size,
                              hipStream_t stream) {
  const float* xyz1  = (const float*)d_in[0];
  const float* xyz2  = (const float*)d_in[1];
  const float* flow1 = (const float*)d_in[2];
  const int*   resol = (const int*)d_in[3];
  float4* ws  = (float4*)d_ws;     // needs PW_B*PW_N1*32 bytes = 512 KB
  float*  out = (float*)d_out;

  pw2_prep<<<(PW_B * PW_N1 + 255) / 256, 256, 0, stream>>>(xyz1, flow1, resol, ws);

  const int tiles = PW_B * (PW_N2 / 16);          // 1024 wave-tiles
  pw2_main<<<tiles / 4, 128, 0, stream>>>(xyz2, resol, ws, out);
}